// GCN_90975997263962
// MI455X (gfx1250) — compile-verified
//
#include <hip/hip_runtime.h>
#include <hip/hip_bf16.h>

typedef __attribute__((ext_vector_type(2))) float v2f;
typedef __attribute__((ext_vector_type(8))) float v8f;

#define TPB 256

// ---------------- utility kernels ----------------

__global__ void gcn_zero(float* __restrict__ p, int n) {
  int i = blockIdx.x * blockDim.x + threadIdx.x;
  if (i < n) p[i] = 0.0f;
}

// deg[dst[e]] += 1  (self loops folded in analytically later)
__global__ void gcn_deg(const int* __restrict__ dst, float* __restrict__ deg, int E) {
  int e = blockIdx.x * blockDim.x + threadIdx.x;
  if (e < E) atomicAdd(&deg[dst[e]], 1.0f);
}

// dinv[n] = rsqrt(deg[n] + 1)   (in place; +1 = self loop)
__global__ void gcn_dinv(float* __restrict__ d, int N) {
  int n = blockIdx.x * blockDim.x + threadIdx.x;
  if (n < N) d[n] = rsqrtf(d[n] + 1.0f);
}

// ---------------- GEMM1: hs = dinv * (X @ W1), agg = dinv * hs  ----------------
// X: [N,128], W1: [128,16]. One wave per 16-node tile, 32x V_WMMA_F32_16X16X4_F32.
__global__ void gcn_gemm1_wmma(const float* __restrict__ X, const float* __restrict__ W1,
                               const float* __restrict__ dinv,
                               float* __restrict__ hs, float* __restrict__ agg, int N) {
  int wave = (blockIdx.x * blockDim.x + threadIdx.x) >> 5;
  int lane = threadIdx.x & 31;
  int row0 = wave * 16;
  if (row0 >= N) return;              // wave-uniform exit (EXEC all-1s for WMMA)
  int m = lane & 15;                  // M row (A) / N col (B,C,D)
  int h = lane >> 4;                  // lane-half selects K pair / row-half
  int arow = row0 + m; if (arow > N - 1) arow = N - 1;   // clamp OOB reads
  const float* xr = X + (size_t)arow * 128;

  v8f c = {};
#pragma unroll 8
  for (int kk = 0; kk < 32; ++kk) {
    int k = kk * 4 + 2 * h;           // A VGPR0=K{0,2}, VGPR1=K{1,3} per lane-half
    v2f a, b;
    a.x = xr[k];
    a.y = xr[k + 1];
    b.x = W1[(size_t)k * 16 + m];     // B row K striped across lanes, same K split
    b.y = W1[(size_t)(k + 1) * 16 + m];
    c = __builtin_amdgcn_wmma_f32_16x16x4_f32(false, a, false, b, (short)0, c, false, false);
  }
#pragma unroll
  for (int r = 0; r < 8; ++r) {       // D: VGPR r -> rows r and r+8 (by lane-half)
    int row = row0 + r + 8 * h;
    if (row < N) {
      float di = dinv[row];
      float t = c[r] * di;            // hs = dinv * (XW1)
      hs[(size_t)row * 16 + m]  = t;
      agg[(size_t)row * 16 + m] = t * di;   // self-loop: dinv^2 * h
    }
  }
}

// ---------------- edge scatter: agg[dst] += dinv[dst] * hs[src] (16 feats) ----------------
// One thread per (edge, 4-feature group): float4 gather + 4 f32 atomics.
__global__ void gcn_scatter(const int* __restrict__ src, const int* __restrict__ dst,
                            const float* __restrict__ dinv, const float* __restrict__ hs,
                            float* __restrict__ agg, int E4) {
  int gid = blockIdx.x * blockDim.x + threadIdx.x;
  if (gid >= E4) return;
  int e = gid >> 2;
  int q = (gid & 3) * 4;
  int s = src[e];
  int d = dst[e];
  float dd = dinv[d];
  float4 v = *(const float4*)(hs + (size_t)s * 16 + q);
  float* ap = agg + (size_t)d * 16 + q;
  atomicAdd(ap + 0, dd * v.x);
  atomicAdd(ap + 1, dd * v.y);
  atomicAdd(ap + 2, dd * v.z);
  atomicAdd(ap + 3, dd * v.w);
}

// ---------------- h = ReLU(agg1 + b1); g = dinv*h; agg2 = dinv*g (in place) ----------------
__global__ void gcn_relu_scale(const float* __restrict__ b1, const float* __restrict__ dinv,
                               float* __restrict__ g /*was hs*/, float* __restrict__ agg /*agg1->agg2*/,
                               int NF) {
  int gid = blockIdx.x * blockDim.x + threadIdx.x;
  if (gid >= NF) return;
  int n = gid >> 4;
  int f = gid & 15;
  float v = agg[gid] + b1[f];
  v = v > 0.0f ? v : 0.0f;            // ReLU
  float di = dinv[n];
  float gg = di * v;
  g[gid]   = gg;                      // pre-scaled for edge gather
  agg[gid] = di * gg;                 // self-loop init for layer-2 aggregation
}

// ---------------- GEMM2: out = agg2 @ W2 + b2 ----------------
// agg2: [N,16], W2: [16,128]. One wave per (16-node tile, 16-col tile): 4x WMMA.
__global__ void gcn_gemm2_wmma(const float* __restrict__ agg2, const float* __restrict__ W2,
                               const float* __restrict__ b2, float* __restrict__ out, int N) {
  int task = (blockIdx.x * blockDim.x + threadIdx.x) >> 5;
  int lane = threadIdx.x & 31;
  int tile = task >> 3;
  int ct = task & 7;
  int row0 = tile * 16;
  if (row0 >= N) return;
  int m = lane & 15;
  int h = lane >> 4;
  int arow = row0 + m; if (arow > N - 1) arow = N - 1;
  const float* ar = agg2 + (size_t)arow * 16;
  int col = ct * 16 + m;

  v8f c = {};
#pragma unroll
  for (int kk = 0; kk < 4; ++kk) {
    int k = kk * 4 + 2 * h;
    v2f a, b;
    a.x = ar[k];
    a.y = ar[k + 1];
    b.x = W2[(size_t)k * 128 + col];
    b.y = W2[(size_t)(k + 1) * 128 + col];
    c = __builtin_amdgcn_wmma_f32_16x16x4_f32(false, a, false, b, (short)0, c, false, false);
  }
  float bias = b2[col];
#pragma unroll
  for (int r = 0; r < 8; ++r) {
    int row = row0 + r + 8 * h;
    if (row < N) out[(size_t)row * 128 + col] = c[r] + bias;
  }
}

// ---------------- launch ----------------

extern "C" void kernel_launch(void* const* d_in, const int* in_sizes, int n_in,
                              void* d_out, int out_size, void* d_ws, size_t ws_size,
                              hipStream_t stream) {
  const float* X  = (const float*)d_in[0];   // [N,128]
  const int*   ei = (const int*)d_in[1];     // [2,E] (int32: JAX x64 disabled)
  const float* W1 = (const float*)d_in[2];   // [128,16]
  const float* b1 = (const float*)d_in[3];   // [16]
  const float* W2 = (const float*)d_in[4];   // [16,128]
  const float* b2 = (const float*)d_in[5];   // [128]
  float* out = (float*)d_out;                // [N,128]

  const int N = in_sizes[0] / 128;
  const int E = in_sizes[1] / 2;
  const int* src = ei;
  const int* dst = ei + E;

  // workspace: dinv[N] | buf1[N*16] (hs -> g) | buf2[N*16] (agg1 -> agg2)
  float* dinv = (float*)d_ws;
  float* buf1 = dinv + N;
  float* buf2 = buf1 + (size_t)N * 16;

  const int tiles = (N + 15) / 16;
  const int NF = N * 16;
  const int E4 = E * 4;

  // 1) degree (zero, count, rsqrt)
  gcn_zero<<<(N + TPB - 1) / TPB, TPB, 0, stream>>>(dinv, N);
  gcn_deg<<<(E + TPB - 1) / TPB, TPB, 0, stream>>>(dst, dinv, E);
  gcn_dinv<<<(N + TPB - 1) / TPB, TPB, 0, stream>>>(dinv, N);

  // 2) layer 1 GEMM (WMMA) + self-loop init
  gcn_gemm1_wmma<<<(tiles * 32 + TPB - 1) / TPB, TPB, 0, stream>>>(X, W1, dinv, buf1, buf2, N);

  // 3) layer 1 edge aggregation (16-dim)
  gcn_scatter<<<(E4 + TPB - 1) / TPB, TPB, 0, stream>>>(src, dst, dinv, buf1, buf2, E4);

  // 4) bias + ReLU + rescale for layer 2 (aggregation moved before GEMM2: linear op)
  gcn_relu_scale<<<(NF + TPB - 1) / TPB, TPB, 0, stream>>>(b1, dinv, buf1, buf2, NF);

  // 5) layer 2 edge aggregation (still 16-dim — 8x less traffic than post-GEMM)
  gcn_scatter<<<(E4 + TPB - 1) / TPB, TPB, 0, stream>>>(src, dst, dinv, buf1, buf2, E4);

  // 6) layer 2 GEMM (WMMA) + bias
  gcn_gemm2_wmma<<<(tiles * 8 * 32 + TPB - 1) / TPB, TPB, 0, stream>>>(buf2, W2, b2, out, N);
}